// GNN_51041391345664
// MI455X (gfx1250) — compile-verified
//
#include <hip/hip_runtime.h>
#include <hip/hip_bf16.h>

typedef __attribute__((ext_vector_type(2))) float v2f;
typedef __attribute__((ext_vector_type(8))) float v8f;

#define TB 256  // threads per block (8 wave32s)

// ---------------------------------------------------------------- utilities
__global__ void k_fill(float* p, float v, int n) {
    int t = blockIdx.x * blockDim.x + threadIdx.x;
    if (t < n) p[t] = v;
}

__global__ void k_count_deg(const int* __restrict__ col, float* __restrict__ deg, int E) {
    int t = blockIdx.x * blockDim.x + threadIdx.x;
    if (t < E) atomicAdd(&deg[col[t]], 1.0f);
}

__global__ void k_dinv(const float* __restrict__ deg, float* __restrict__ dinv, int n) {
    int t = blockIdx.x * blockDim.x + threadIdx.x;
    if (t < n) {
        float d = deg[t];
        dinv[t] = d > 0.0f ? rsqrtf(d) : 0.0f;
    }
}

__global__ void k_init_bias64(float* __restrict__ agg, const float* __restrict__ b, int total) {
    int t = blockIdx.x * blockDim.x + threadIdx.x;
    if (t < total) agg[t] = b[t & 63];
}

// pad x [N,7] -> x_pad [N,8] (zero in lane 7)
__global__ void k_pad_x(const float* __restrict__ x, float* __restrict__ xp, int N) {
    int t = blockIdx.x * blockDim.x + threadIdx.x;
    if (t >= N * 8) return;
    int i = t >> 3, f = t & 7;
    xp[t] = (f < 7) ? x[i * 7 + f] : 0.0f;
}

// transpose W [K,NCOLS] -> Wt [NCOLS,Kpad] (zero-pad K)
__global__ void k_transpose(const float* __restrict__ W, float* __restrict__ Wt,
                            int K, int Kpad, int NCOLS) {
    int t = blockIdx.x * blockDim.x + threadIdx.x;
    if (t >= NCOLS * Kpad) return;
    int n = t / Kpad, k = t % Kpad;
    Wt[t] = (k < K) ? W[k * NCOLS + n] : 0.0f;
}

// ------------------------------------------------- WMMA fp32 GEMM (M x K x NCOLS)
// C[M,NCOLS] = op(A[M,K]) @ Bt^T  (Bt is [NCOLS,K], i.e. pre-transposed B)
// One wave32 computes a 32-row x NCOLS tile (2 row-tiles) with
// V_WMMA_F32_16X16X4_F32. Fragment layouts per CDNA5 ISA 7.12.2:
//   A 16x4 f32: lanes 0-15 hold {K=k0,k0+1} for row=lane; lanes 16-31 {k0+2,k0+3}.
//   B 4x16 f32: lane L holds col N=L%16, same K split  -> contiguous pair in Bt.
//   C/D: VGPR v -> rows v / v+8, col = lane%16.
template <int K, int NCOLS, bool RELU_A, bool HAS_BIAS>
__global__ void k_gemm_wmma(const float* __restrict__ A, const float* __restrict__ Bt,
                            const float* __restrict__ bias, float* __restrict__ C, int M) {
    constexpr int NT = NCOLS / 16;
    int wave = (int)((blockIdx.x * blockDim.x + threadIdx.x) >> 5);
    int lane = (int)(threadIdx.x & 31);
    int nSuper = M >> 5;                 // 32 rows per wave; M=100000 -> 3125 exact
    if (wave >= nSuper) return;          // wave-uniform exit; EXEC stays all-1s

    int row0 = wave << 5;
    int m    = lane & 15;
    bool hi  = lane >= 16;
    int koff = hi ? 2 : 0;

    v8f acc[2][NT];
    v8f zero = {};
#pragma unroll
    for (int mt = 0; mt < 2; ++mt)
#pragma unroll
        for (int t = 0; t < NT; ++t) acc[mt][t] = zero;

    const float* arow0 = A + (size_t)(row0 + m) * K;     // lda == K (padded inputs)
    const float* arow1 = arow0 + (size_t)16 * K;
    const float* brow  = Bt + (size_t)m * K;             // lane's B column, K-major

#pragma unroll
    for (int k0 = 0; k0 < K; k0 += 4) {
        int ka = k0 + koff;
        v2f a0 = *(const v2f*)(arow0 + ka);
        v2f a1 = *(const v2f*)(arow1 + ka);
        if (RELU_A) {
            a0.x = fmaxf(a0.x, 0.0f); a0.y = fmaxf(a0.y, 0.0f);
            a1.x = fmaxf(a1.x, 0.0f); a1.y = fmaxf(a1.y, 0.0f);
        }
#pragma unroll
        for (int t = 0; t < NT; ++t) {
            v2f b = *(const v2f*)(brow + (size_t)(t * 16) * K + ka);
            acc[0][t] = __builtin_amdgcn_wmma_f32_16x16x4_f32(
                false, a0, false, b, (short)0, acc[0][t], false, false);
            acc[1][t] = __builtin_amdgcn_wmma_f32_16x16x4_f32(
                false, a1, false, b, (short)0, acc[1][t], false, false);
        }
    }

#pragma unroll
    for (int mt = 0; mt < 2; ++mt)
#pragma unroll
        for (int t = 0; t < NT; ++t) {
            int cc = t * 16 + m;
            float bv = 0.0f;
            if (HAS_BIAS) bv = bias[cc];
#pragma unroll
            for (int v = 0; v < 8; ++v) {
                int rr = row0 + mt * 16 + v + (hi ? 8 : 0);
                float val = acc[mt][t][v];
                if (HAS_BIAS) val += bv;          // constexpr-dead when !HAS_BIAS
                C[(size_t)rr * NCOLS + cc] = val;
            }
        }
}

// --------------------------------------------- edge scatter (aggregation), F=64
// One wave32 per edge; lane owns the adjacent feature pair {2*lane, 2*lane+1}:
// one aligned global_load_b64 per lane, two adjacent f32 atomics.
__global__ void k_scatter64(const float* __restrict__ h, const int* __restrict__ row,
                            const int* __restrict__ col, const float* __restrict__ dinv,
                            float* __restrict__ agg, int E, int N) {
    int wave = (int)((blockIdx.x * blockDim.x + threadIdx.x) >> 5);
    int lane = (int)(threadIdx.x & 31);
    int total = E + N;
    if (wave >= total) return;

    int r, c;
    if (wave < E) { r = row[wave]; c = col[wave]; }
    else          { r = wave - E; c = r; }        // self loop

    float norm = dinv[r] * dinv[c];
    v2f hv = *(const v2f*)(h + (size_t)r * 64 + 2 * lane);
    float* ac = agg + (size_t)c * 64 + 2 * lane;
    atomicAdd(&ac[0], hv.x * norm);
    atomicAdd(&ac[1], hv.y * norm);
}

// --------------------------------------------------------------- mean pool
__global__ void k_pool_accum(const float* __restrict__ h3, const int* __restrict__ batch,
                             float* __restrict__ psum, float* __restrict__ pcnt, int N) {
    int t = blockIdx.x * blockDim.x + threadIdx.x;
    if (t >= N * 32) return;
    int i = t >> 5, f = t & 31;
    int g = batch[i];
    atomicAdd(&psum[g * 32 + f], h3[t]);
    if (f == 0) atomicAdd(&pcnt[g], 1.0f);
}

__global__ void k_pool_final(const float* __restrict__ psum, const float* __restrict__ pcnt,
                             float* __restrict__ out, int total) {
    int t = blockIdx.x * blockDim.x + threadIdx.x;
    if (t < total) {
        int g = t >> 5;  // 32 outputs per graph
        out[t] = psum[t] / fmaxf(pcnt[g], 1.0f);
    }
}

static inline int cdiv(long long a, long long b) { return (int)((a + b - 1) / b); }

extern "C" void kernel_launch(void* const* d_in, const int* in_sizes, int n_in,
                              void* d_out, int out_size, void* d_ws, size_t ws_size,
                              hipStream_t stream) {
    const float* x    = (const float*)d_in[0];   // [N,7]
    const int*   eidx = (const int*)d_in[1];     // [2,E] flat
    const int*   batch= (const int*)d_in[2];     // [N]
    const float* W1   = (const float*)d_in[3];   // [7,64]
    const float* b1   = (const float*)d_in[4];   // [64]
    const float* W2   = (const float*)d_in[5];   // [64,64]
    const float* b2   = (const float*)d_in[6];   // [64]
    const float* Wlin = (const float*)d_in[7];   // [64,32]
    const float* blin = (const float*)d_in[8];   // [32]
    float* out = (float*)d_out;                  // [64,32]

    const int N = in_sizes[2];        // 100000
    const int E = in_sizes[1] / 2;    // 1600000
    const int* erow = eidx;
    const int* ecol = eidx + E;

    // workspace layout (floats; all offsets even -> b64 loads stay 8B-aligned)
    float* p    = (float*)d_ws;
    float* deg  = p;                               // N
    float* dinv = deg  + (size_t)N;                // N
    float* xpad = dinv + (size_t)N;                // N*8   (x zero-padded to K=8)
    float* bufA = xpad + (size_t)N * 8;            // N*64  (h1 / h2 / h3)
    float* bufB = bufA + (size_t)N * 64;           // N*64  (agg1 / agg2)
    float* Wt1  = bufB + (size_t)N * 64;           // 64*8   (W1^T, K padded 7->8)
    float* Wt2  = Wt1  + 64 * 8;                   // 64*64  (W2^T)
    float* Wtl  = Wt2  + 64 * 64;                  // 32*64  (Wlin^T)
    float* psum = Wtl  + 32 * 64;                  // 64*32
    float* pcnt = psum + 64 * 32;                  // 64

    const int gemmBl = cdiv(N >> 5, TB / 32);      // 3125 waves, 8 per block
    const int edgeBl = cdiv((long long)E + N, TB / 32);

    // ---- setup: degrees, padded x, transposed weights
    k_fill<<<cdiv(N, TB), TB, 0, stream>>>(deg, 1.0f, N);
    k_count_deg<<<cdiv(E, TB), TB, 0, stream>>>(ecol, deg, E);
    k_dinv<<<cdiv(N, TB), TB, 0, stream>>>(deg, dinv, N);
    k_pad_x<<<cdiv((long long)N * 8, TB), TB, 0, stream>>>(x, xpad, N);
    k_transpose<<<cdiv(64 * 8,  TB), TB, 0, stream>>>(W1,   Wt1, 7,  8,  64);
    k_transpose<<<cdiv(64 * 64, TB), TB, 0, stream>>>(W2,   Wt2, 64, 64, 64);
    k_transpose<<<cdiv(32 * 64, TB), TB, 0, stream>>>(Wlin, Wtl, 64, 64, 32);

    // ---- layer 1: h1 = x @ W1 ; agg1 = b1 + scatter(norm * h1[row])
    k_gemm_wmma<8, 64, false, false><<<gemmBl, TB, 0, stream>>>(xpad, Wt1, nullptr, bufA, N);
    k_init_bias64<<<cdiv((long long)N * 64, TB), TB, 0, stream>>>(bufB, b1, N * 64);
    k_scatter64<<<edgeBl, TB, 0, stream>>>(bufA, erow, ecol, dinv, bufB, E, N);

    // ---- layer 2: h2 = relu(agg1) @ W2 ; agg2 = b2 + scatter(norm * h2[row])
    k_gemm_wmma<64, 64, true, false><<<gemmBl, TB, 0, stream>>>(bufB, Wt2, nullptr, bufA, N);
    k_init_bias64<<<cdiv((long long)N * 64, TB), TB, 0, stream>>>(bufB, b2, N * 64);
    k_scatter64<<<edgeBl, TB, 0, stream>>>(bufA, erow, ecol, dinv, bufB, E, N);

    // ---- head: h3 = relu(agg2) @ Wlin + blin
    k_gemm_wmma<64, 32, true, true><<<gemmBl, TB, 0, stream>>>(bufB, Wtl, blin, bufA, N);

    // ---- global mean pool
    k_fill<<<cdiv(64 * 32 + 64, TB), TB, 0, stream>>>(psum, 0.0f, 64 * 32 + 64);
    k_pool_accum<<<cdiv((long long)N * 32, TB), TB, 0, stream>>>(bufA, batch, psum, pcnt, N);
    k_pool_final<<<cdiv(64 * 32, TB), TB, 0, stream>>>(psum, pcnt, out, 64 * 32);
}